// HybridLayer_78700980732390
// MI455X (gfx1250) — compile-verified
//
#include <hip/hip_runtime.h>
#include <hip/hip_bf16.h>

typedef __attribute__((ext_vector_type(2))) float v2f;
typedef __attribute__((ext_vector_type(8))) float v8f;

#define NQ       10
#define DIMQ     1024      // 2^NQ
#define KDIM     512       // IN_FEATURES
#define OUTF     64
#define NBATCH   4096
#define NLAYERS  4

// ---------------------------------------------------------------------------
// Kernel 1: precompute the 40 Rot(phi,theta,omega) 2x2 complex unitaries.
// U[g] = [[e^{-i(phi+om)/2} c, -e^{+i(phi-om)/2} s],
//         [e^{-i(phi-om)/2} s,  e^{+i(phi+om)/2} c]]
// Stored as 8 floats per gate: u00r,u00i,u01r,u01i,u10r,u10i,u11r,u11i.
// ---------------------------------------------------------------------------
__global__ void rot_prep_kernel(const float* __restrict__ qw,
                                float* __restrict__ U) {
  int g = blockIdx.x * blockDim.x + threadIdx.x;
  if (g >= NLAYERS * NQ) return;
  float phi = qw[3 * g + 0];
  float th  = qw[3 * g + 1];
  float om  = qw[3 * g + 2];
  float ch = cosf(0.5f * th), sh = sinf(0.5f * th);
  float a  = 0.5f * (phi + om), b = 0.5f * (phi - om);
  float ca = cosf(a), sa = sinf(a);
  float cb = cosf(b), sb = sinf(b);
  float* u = U + 8 * g;
  u[0] =  ch * ca;  u[1] = -ch * sa;   // e_mm * c
  u[2] = -sh * cb;  u[3] = -sh * sb;   // -e_pm * s
  u[4] =  sh * cb;  u[5] = -sh * sb;   // e_mp * s
  u[6] =  ch * ca;  u[7] =  ch * sa;   // e_pp * c
}

// ---------------------------------------------------------------------------
// Kernel 2: H[4096,1024] = relu(X[4096,512] @ W[1024,512]^T + bias)
// f32 WMMA 16x16x4. 8 waves/block; wave tile 32x32; block tile 128x64.
// A frag (16x4): lane m = lane%16, K pair = (lane/16)*2 -> float2 per lane.
// B frag (4x16): lane n = lane%16, same K pairing; B[k][n] = W[n][k] so
// contiguous-K float2 loads from row-major W.
// C 16x16: VGPR r, lanes 0-15 -> M=r, lanes 16-31 -> M=r+8; N = lane%16.
// ---------------------------------------------------------------------------
__global__ __launch_bounds__(256) void gemm_relu_kernel(
    const float* __restrict__ X,     // [4096,512]
    const float* __restrict__ W,     // [1024,512]
    const float* __restrict__ bias,  // [1024]
    float* __restrict__ H)           // [4096,1024]
{
  const int lane = threadIdx.x & 31;
  const int wave = threadIdx.x >> 5;
  const int wm = wave >> 1;            // 0..3
  const int wn = wave & 1;             // 0..1
  const int m_base = blockIdx.x * 128 + wm * 32;
  const int n_base = blockIdx.y * 64  + wn * 32;

  const int r16  = lane & 15;          // row (A) / col (B) within tile
  const int half = lane >> 4;          // K-pair selector

  v8f acc00 = {}; v8f acc01 = {}; v8f acc10 = {}; v8f acc11 = {};

  const float* aptr0 = X + (size_t)(m_base + r16) * KDIM + 2 * half;
  const float* aptr1 = aptr0 + (size_t)16 * KDIM;
  const float* bptr0 = W + (size_t)(n_base + r16) * KDIM + 2 * half;
  const float* bptr1 = bptr0 + (size_t)16 * KDIM;

#pragma unroll 4
  for (int k = 0; k < KDIM; k += 4) {
    v2f a0 = *(const v2f*)(aptr0 + k);
    v2f a1 = *(const v2f*)(aptr1 + k);
    v2f b0 = *(const v2f*)(bptr0 + k);
    v2f b1 = *(const v2f*)(bptr1 + k);
    acc00 = __builtin_amdgcn_wmma_f32_16x16x4_f32(false, a0, false, b0,
                                                  (short)0, acc00, false, false);
    acc01 = __builtin_amdgcn_wmma_f32_16x16x4_f32(false, a0, false, b1,
                                                  (short)0, acc01, false, false);
    acc10 = __builtin_amdgcn_wmma_f32_16x16x4_f32(false, a1, false, b0,
                                                  (short)0, acc10, false, false);
    acc11 = __builtin_amdgcn_wmma_f32_16x16x4_f32(false, a1, false, b1,
                                                  (short)0, acc11, false, false);
  }

  const int row_off = half << 3;  // lanes 16-31 hold M = r + 8
#pragma unroll
  for (int i = 0; i < 2; ++i) {
#pragma unroll
    for (int j = 0; j < 2; ++j) {
      const v8f acc = (i == 0) ? (j == 0 ? acc00 : acc01)
                               : (j == 0 ? acc10 : acc11);
      const int col = n_base + j * 16 + r16;
      const float bv = bias[col];
#pragma unroll
      for (int r = 0; r < 8; ++r) {
        const int row = m_base + i * 16 + row_off + r;
        float v = acc[r] + bv;
        H[(size_t)row * DIMQ + col] = fmaxf(v, 0.0f);
      }
    }
  }
}

// ---------------------------------------------------------------------------
// Kernel 3: per-batch 10-qubit statevector simulation + PauliZ expvals +
// final linear layer. One block (256 threads, 8 waves) per batch element;
// state lives in LDS (float2[1024] = 8 KB).
// Wire w maps to bit (9 - w) of the flat amplitude index.
// ---------------------------------------------------------------------------
__global__ __launch_bounds__(256) void qsim_kernel(
    const float* __restrict__ H,      // [4096,1024] (post-ReLU)
    const float* __restrict__ U,      // [40*8] rot matrices
    const float* __restrict__ Wpost,  // [64,10]
    const float* __restrict__ bpost,  // [64]
    float* __restrict__ out)          // [4096,64]
{
  __shared__ float2 psi[DIMQ];
  __shared__ float  uS[NLAYERS * NQ * 8];
  __shared__ float  zS[NQ];
  __shared__ float  ssum;

  const int tid = threadIdx.x;
  const int b   = blockIdx.x;

  for (int i = tid; i < NLAYERS * NQ * 8; i += 256) uS[i] = U[i];
  if (tid == 0) ssum = 0.0f;
  if (tid < NQ) zS[tid] = 0.0f;
  __syncthreads();

  // Normalize amplitudes (AmplitudeEmbedding, normalize=True).
  const float* hrow = H + (size_t)b * DIMQ;
  float v0 = hrow[tid], v1 = hrow[tid + 256], v2 = hrow[tid + 512], v3 = hrow[tid + 768];
  atomicAdd(&ssum, v0 * v0 + v1 * v1 + v2 * v2 + v3 * v3);
  __syncthreads();
  const float inv = rsqrtf(ssum);
  psi[tid      ] = make_float2(v0 * inv, 0.0f);
  psi[tid + 256] = make_float2(v1 * inv, 0.0f);
  psi[tid + 512] = make_float2(v2 * inv, 0.0f);
  psi[tid + 768] = make_float2(v3 * inv, 0.0f);

  // StronglyEntanglingLayers
  for (int l = 0; l < NLAYERS; ++l) {
    for (int w = 0; w < NQ; ++w) {
      __syncthreads();
      const float* u = &uS[(l * NQ + w) * 8];
      const float u00r = u[0], u00i = u[1], u01r = u[2], u01i = u[3];
      const float u10r = u[4], u10i = u[5], u11r = u[6], u11i = u[7];
      const int mask = 1 << (9 - w);
#pragma unroll
      for (int pp = 0; pp < 2; ++pp) {
        const int p  = tid + pp * 256;                         // pair id 0..511
        const int i0 = ((p & ~(mask - 1)) << 1) | (p & (mask - 1));
        const int i1 = i0 | mask;
        const float2 a0 = psi[i0];
        const float2 a1 = psi[i1];
        float n0r = u00r * a0.x - u00i * a0.y + u01r * a1.x - u01i * a1.y;
        float n0i = u00r * a0.y + u00i * a0.x + u01r * a1.y + u01i * a1.x;
        float n1r = u10r * a0.x - u10i * a0.y + u11r * a1.x - u11i * a1.y;
        float n1i = u10r * a0.y + u10i * a0.x + u11r * a1.y + u11i * a1.x;
        psi[i0] = make_float2(n0r, n0i);
        psi[i1] = make_float2(n1r, n1i);
      }
    }
    const int r = (l % (NQ - 1)) + 1;   // PennyLane default ranges
    for (int w = 0; w < NQ; ++w) {
      __syncthreads();
      const int t  = (w + r) % NQ;
      const int pc = 9 - w;             // control bit position
      const int pt = 9 - t;             // target bit position
      const int p1 = pc < pt ? pc : pt;
      const int p2 = pc < pt ? pt : pc;
      int x = tid;                      // 8 free bits
      x = ((x & ~((1 << p1) - 1)) << 1) | (x & ((1 << p1) - 1));
      x = ((x & ~((1 << p2) - 1)) << 1) | (x & ((1 << p2) - 1));
      const int i0 = x | (1 << pc);     // control=1, target=0
      const int i1 = i0 | (1 << pt);    // control=1, target=1
      const float2 t0 = psi[i0];
      const float2 t1 = psi[i1];
      psi[i0] = t1;
      psi[i1] = t0;
    }
  }
  __syncthreads();

  // PauliZ expectation values.
  float zl[NQ];
#pragma unroll
  for (int q = 0; q < NQ; ++q) zl[q] = 0.0f;
#pragma unroll
  for (int j = 0; j < 4; ++j) {
    const int i = tid + 256 * j;
    const float2 a = psi[i];
    const float pr = a.x * a.x + a.y * a.y;
#pragma unroll
    for (int q = 0; q < NQ; ++q)
      zl[q] += ((i >> (9 - q)) & 1) ? -pr : pr;
  }
#pragma unroll
  for (int q = 0; q < NQ; ++q) atomicAdd(&zS[q], zl[q]);
  __syncthreads();

  // Final linear layer: out[b,o] = sum_q z[q] * Wpost[o,q] + bpost[o]
  if (tid < OUTF) {
    float acc = bpost[tid];
#pragma unroll
    for (int q = 0; q < NQ; ++q) acc += zS[q] * Wpost[tid * NQ + q];
    out[(size_t)b * OUTF + tid] = acc;
  }
}

// ---------------------------------------------------------------------------
extern "C" void kernel_launch(void* const* d_in, const int* in_sizes, int n_in,
                              void* d_out, int out_size, void* d_ws, size_t ws_size,
                              hipStream_t stream) {
  const float* x      = (const float*)d_in[0];  // [4096,512]
  const float* W_pre  = (const float*)d_in[1];  // [1024,512]
  const float* b_pre  = (const float*)d_in[2];  // [1024]
  const float* qw     = (const float*)d_in[3];  // [4,10,3]
  const float* W_post = (const float*)d_in[4];  // [64,10]
  const float* b_post = (const float*)d_in[5];  // [64]
  float* out = (float*)d_out;                   // [4096,64]

  float* U = (float*)d_ws;                                  // 320 floats
  float* H = (float*)((char*)d_ws + 4096);                  // 4096x1024 f32

  rot_prep_kernel<<<1, 64, 0, stream>>>(qw, U);
  gemm_relu_kernel<<<dim3(NBATCH / 128, DIMQ / 64), 256, 0, stream>>>(x, W_pre, b_pre, H);
  qsim_kernel<<<NBATCH, 256, 0, stream>>>(H, U, W_post, b_post, out);
}